// HMM_31447750542094
// MI455X (gfx1250) — compile-verified
//
#include <hip/hip_runtime.h>
#include <hip/hip_bf16.h>

// HMM forward-backward, B=64, N=64, T=4096.
// Strategy: exp-domain rescaled recurrences -> per-step 64x64x64 f32 GEMM via
// V_WMMA_F32_16X16X4_F32. One wave owns 16 batches x 64 states (4 C tiles).
// 4 single-wave workgroups per direction. gamma fused into the backward scan.

#define HMM_B 64
#define HMM_N 64
#define HMM_T 4096
#define LDS_RS 68   // padded row stride (dwords) for the 16x64 P transpose

typedef __attribute__((ext_vector_type(2))) float v2f;
typedef __attribute__((ext_vector_type(8))) float v8f;

__device__ __forceinline__ float half_reduce_max(float v) {
  // butterfly within each 16-lane half (batch rows live per-half)
  v = fmaxf(v, __shfl_xor(v, 1, 32));
  v = fmaxf(v, __shfl_xor(v, 2, 32));
  v = fmaxf(v, __shfl_xor(v, 4, 32));
  v = fmaxf(v, __shfl_xor(v, 8, 32));
  return v;
}
__device__ __forceinline__ float half_reduce_sum(float v) {
  v += __shfl_xor(v, 1, 32);
  v += __shfl_xor(v, 2, 32);
  v += __shfl_xor(v, 4, 32);
  v += __shfl_xor(v, 8, 32);
  return v;
}

template <bool TR>
__device__ __forceinline__ float load_em(const float* __restrict__ em,
                                         const float* __restrict__ emT,
                                         int t, int b, int n) {
  if (TR) return emT[(size_t)t * (HMM_B * HMM_N) + b * HMM_N + n];
  return em[((size_t)b * HMM_N + n) * HMM_T + t];
}

// ---------------- emissions transpose: [B*N, T] -> [T, B*N] ----------------
__global__ __launch_bounds__(256) void hmm_transpose_em(
    const float* __restrict__ src, float* __restrict__ dst) {
  __shared__ float tile[32][33];
  const int tx = threadIdx.x, ty = threadIdx.y;          // 32 x 8
  const int x0 = blockIdx.x * 32;                        // t
  const int y0 = blockIdx.y * 32;                        // b*N+n
#pragma unroll
  for (int i = 0; i < 32; i += 8)
    tile[ty + i][tx] = src[(size_t)(y0 + ty + i) * HMM_T + (x0 + tx)];
  __syncthreads();
#pragma unroll
  for (int i = 0; i < 32; i += 8)
    dst[(size_t)(x0 + ty + i) * (HMM_B * HMM_N) + (y0 + tx)] = tile[tx][ty + i];
}

// ---------------- forward scan: writes log_alpha (strided) into gbuf --------
template <bool TR>
__global__ __launch_bounds__(32) void hmm_forward(
    const float* __restrict__ em, const float* __restrict__ emT,
    const float* __restrict__ init_state, const float* __restrict__ tm,
    float* __restrict__ gbuf) {
  __shared__ float plds[16 * LDS_RS];
  const int lane = threadIdx.x & 31;
  const int lo = lane & 15, hi = lane >> 4;
  const int b0 = blockIdx.x * 16;

  // B fragments (loop-invariant): B[k=i][n=j] = tm[i][j]
  v2f bfrag[4][16];
#pragma unroll
  for (int nb = 0; nb < 4; ++nb)
#pragma unroll
    for (int kb = 0; kb < 16; ++kb) {
      const int k = 4 * kb + 2 * hi, n = 16 * nb + lo;
      bfrag[nb][kb].x = tm[k * HMM_N + n];
      bfrag[nb][kb].y = tm[(k + 1) * HMM_N + n];
    }

  // alpha_0 = log(init) + e_0, in C layout: a[nb][r] -> b=b0+r+8*hi, n=16nb+lo
  float a_cur[4][8];
  float linit[4];
#pragma unroll
  for (int nb = 0; nb < 4; ++nb) linit[nb] = __logf(init_state[16 * nb + lo]);
#pragma unroll
  for (int nb = 0; nb < 4; ++nb)
#pragma unroll
    for (int r = 0; r < 8; ++r) {
      const int b = b0 + r + 8 * hi, n = 16 * nb + lo;
      const float a0 = linit[nb] + load_em<TR>(em, emT, 0, b, n);
      a_cur[nb][r] = a0;
      gbuf[((size_t)b * HMM_N + n) * HMM_T + 0] = a0;
    }

#pragma unroll 1
  for (int t = 1; t < HMM_T; ++t) {
    // per-batch max over the 64 states
    float m[8];
#pragma unroll
    for (int r = 0; r < 8; ++r) {
      float v = fmaxf(fmaxf(a_cur[0][r], a_cur[1][r]),
                      fmaxf(a_cur[2][r], a_cur[3][r]));
      m[r] = half_reduce_max(v);
    }
    // P = exp(a - m) -> LDS, layout [b_local][state], padded row
#pragma unroll
    for (int nb = 0; nb < 4; ++nb)
#pragma unroll
      for (int r = 0; r < 8; ++r)
        plds[(r + 8 * hi) * LDS_RS + nb * 16 + lo] =
            __expf(a_cur[nb][r] - m[r]);
    __syncthreads();
    v2f af[16];
#pragma unroll
    for (int kb = 0; kb < 16; ++kb)
      af[kb] = *(const v2f*)&plds[lo * LDS_RS + 4 * kb + 2 * hi];
    __syncthreads();

    // S[b, j] = sum_i P[b, i] * tm[i][j]
    v8f c[4];
#pragma unroll
    for (int nb = 0; nb < 4; ++nb)
#pragma unroll
      for (int r = 0; r < 8; ++r) c[nb][r] = 0.0f;
#pragma unroll
    for (int kb = 0; kb < 16; ++kb)
#pragma unroll
      for (int nb = 0; nb < 4; ++nb)
        c[nb] = __builtin_amdgcn_wmma_f32_16x16x4_f32(
            false, af[kb], false, bfrag[nb][kb], (short)0, c[nb], false, false);

    // alpha_t = m + log(S) + e_t ; store strided
#pragma unroll
    for (int nb = 0; nb < 4; ++nb)
#pragma unroll
      for (int r = 0; r < 8; ++r) {
        const int b = b0 + r + 8 * hi, n = 16 * nb + lo;
        const float na =
            m[r] + __logf(c[nb][r]) + load_em<TR>(em, emT, t, b, n);
        a_cur[nb][r] = na;
        gbuf[((size_t)b * HMM_N + n) * HMM_T + t] = na;
      }
  }
}

// ------------- backward scan with fused gamma (in-place over gbuf) ----------
template <bool TR>
__global__ __launch_bounds__(32) void hmm_backward_gamma(
    const float* __restrict__ em, const float* __restrict__ emT,
    const float* __restrict__ tm, float* __restrict__ gbuf) {
  __shared__ float plds[16 * LDS_RS];
  const int lane = threadIdx.x & 31;
  const int lo = lane & 15, hi = lane >> 4;
  const int b0 = blockIdx.x * 16;

  // B fragments: S[b, i] = sum_j W[b, j] * tm[i][j] -> B[k=j][n=i] = tm[i][j]
  v2f bfrag[4][16];
#pragma unroll
  for (int nb = 0; nb < 4; ++nb)
#pragma unroll
    for (int kb = 0; kb < 16; ++kb) {
      const int k = 4 * kb + 2 * hi, n = 16 * nb + lo;
      bfrag[nb][kb].x = tm[n * HMM_N + k];
      bfrag[nb][kb].y = tm[n * HMM_N + k + 1];
    }

  float beta[4][8];
#pragma unroll
  for (int nb = 0; nb < 4; ++nb)
#pragma unroll
    for (int r = 0; r < 8; ++r) beta[nb][r] = 0.0f;

  // gamma emit: g = alpha_t (from gbuf) + beta_t, normalize over states
  auto emit_gamma = [&](int t, float bt[4][8]) {
    float g[4][8];
#pragma unroll
    for (int nb = 0; nb < 4; ++nb)
#pragma unroll
      for (int r = 0; r < 8; ++r) {
        const int b = b0 + r + 8 * hi, n = 16 * nb + lo;
        g[nb][r] = gbuf[((size_t)b * HMM_N + n) * HMM_T + t] + bt[nb][r];
      }
#pragma unroll
    for (int r = 0; r < 8; ++r) {
      float mv = fmaxf(fmaxf(g[0][r], g[1][r]), fmaxf(g[2][r], g[3][r]));
      mv = half_reduce_max(mv);
      float s = __expf(g[0][r] - mv) + __expf(g[1][r] - mv) +
                __expf(g[2][r] - mv) + __expf(g[3][r] - mv);
      s = half_reduce_sum(s);
      const float lse = mv + __logf(s);
#pragma unroll
      for (int nb = 0; nb < 4; ++nb) {
        const int b = b0 + r + 8 * hi, n = 16 * nb + lo;
        gbuf[((size_t)b * HMM_N + n) * HMM_T + t] = g[nb][r] - lse;
      }
    }
  };

  emit_gamma(HMM_T - 1, beta);  // beta_{T-1} = 0

#pragma unroll 1
  for (int t = HMM_T - 2; t >= 0; --t) {
    // w = e_{t+1} + beta_{t+1}; rescale and exponentiate
    float w[4][8], m[8];
#pragma unroll
    for (int nb = 0; nb < 4; ++nb)
#pragma unroll
      for (int r = 0; r < 8; ++r) {
        const int b = b0 + r + 8 * hi, n = 16 * nb + lo;
        w[nb][r] = load_em<TR>(em, emT, t + 1, b, n) + beta[nb][r];
      }
#pragma unroll
    for (int r = 0; r < 8; ++r) {
      float v = fmaxf(fmaxf(w[0][r], w[1][r]), fmaxf(w[2][r], w[3][r]));
      m[r] = half_reduce_max(v);
    }
#pragma unroll
    for (int nb = 0; nb < 4; ++nb)
#pragma unroll
      for (int r = 0; r < 8; ++r)
        plds[(r + 8 * hi) * LDS_RS + nb * 16 + lo] = __expf(w[nb][r] - m[r]);
    __syncthreads();
    v2f af[16];
#pragma unroll
    for (int kb = 0; kb < 16; ++kb)
      af[kb] = *(const v2f*)&plds[lo * LDS_RS + 4 * kb + 2 * hi];
    __syncthreads();

    v8f c[4];
#pragma unroll
    for (int nb = 0; nb < 4; ++nb)
#pragma unroll
      for (int r = 0; r < 8; ++r) c[nb][r] = 0.0f;
#pragma unroll
    for (int kb = 0; kb < 16; ++kb)
#pragma unroll
      for (int nb = 0; nb < 4; ++nb)
        c[nb] = __builtin_amdgcn_wmma_f32_16x16x4_f32(
            false, af[kb], false, bfrag[nb][kb], (short)0, c[nb], false, false);

    // beta_t = m + log(S); S in (0, 1] since tm rows are stochastic
#pragma unroll
    for (int nb = 0; nb < 4; ++nb)
#pragma unroll
      for (int r = 0; r < 8; ++r) beta[nb][r] = m[r] + __logf(c[nb][r]);

    emit_gamma(t, beta);
  }
}

extern "C" void kernel_launch(void* const* d_in, const int* in_sizes, int n_in,
                              void* d_out, int out_size, void* d_ws,
                              size_t ws_size, hipStream_t stream) {
  (void)in_sizes; (void)n_in; (void)out_size;
  const float* em = (const float*)d_in[0];          // [B, N, T]
  const float* init_state = (const float*)d_in[1];  // [N]
  const float* tm = (const float*)d_in[2];          // [N, N]
  float* out = (float*)d_out;                       // [B, N, T]

  const size_t emt_bytes = (size_t)HMM_B * HMM_N * HMM_T * sizeof(float);
  if (ws_size >= emt_bytes) {
    float* emT = (float*)d_ws;  // [T, B, N]
    hmm_transpose_em<<<dim3(HMM_T / 32, (HMM_B * HMM_N) / 32), dim3(32, 8), 0,
                       stream>>>(em, emT);
    hmm_forward<true><<<HMM_B / 16, 32, 0, stream>>>(em, emT, init_state, tm,
                                                     out);
    hmm_backward_gamma<true><<<HMM_B / 16, 32, 0, stream>>>(em, emT, tm, out);
  } else {
    hmm_forward<false><<<HMM_B / 16, 32, 0, stream>>>(em, nullptr, init_state,
                                                      tm, out);
    hmm_backward_gamma<false><<<HMM_B / 16, 32, 0, stream>>>(em, nullptr, tm,
                                                             out);
  }
}